// LinearChanneled_57939108823730
// MI455X (gfx1250) — compile-verified
//
#include <hip/hip_runtime.h>

// Problem constants (from reference)
#define IN_CH   8
#define OUT_CH  8
#define IN_F    512
#define OUT_F   512
#define BATCH   64
#define K_TOT   (IN_CH * IN_F)     // 4096  (k = c*512 + i)
#define M_TOT   (OUT_CH * OUT_F)   // 4096  (m = g*512 + o)
#define GOI     (OUT_CH * OUT_F * IN_F)   // per-c stride in W / bias = 2,097,152

typedef __attribute__((ext_vector_type(2))) float v2f;
typedef __attribute__((ext_vector_type(8))) float v8f;

// ---------------------------------------------------------------------------
// Kernel 1: b_red[m] = mean over (c, i) of bias[c, g, o, i], m = g*512 + o.
// One 256-thread block per m; coalesced NT streaming of bias (64 MB, one use).
// ---------------------------------------------------------------------------
__global__ __launch_bounds__(256)
void bias_reduce_kernel(const float* __restrict__ bias, float* __restrict__ bred) {
    const int m = blockIdx.x;        // 0..4095
    const int t = threadIdx.x;       // 0..255
    const float* base = bias + (size_t)m * IN_F;
    float s = 0.0f;
#pragma unroll
    for (int c = 0; c < IN_CH; ++c) {
        const float* pc = base + (size_t)c * GOI;
        s += __builtin_nontemporal_load(pc + t) +
             __builtin_nontemporal_load(pc + t + 256);
    }
    __shared__ float sm[256];
    sm[t] = s;
    __syncthreads();
    for (int off = 128; off > 0; off >>= 1) {
        if (t < off) sm[t] += sm[t + off];
        __syncthreads();
    }
    if (t == 0) bred[m] = sm[0] * (1.0f / (float)(IN_CH * IN_F));
}

// ---------------------------------------------------------------------------
// Kernel 2: out[b, o, g] = b_red[g*512 + o]   (GEMM accumulates on top)
// out flat index: b*4096 + o*8 + g
// ---------------------------------------------------------------------------
__global__ __launch_bounds__(256)
void out_init_kernel(const float* __restrict__ bred, float* __restrict__ out) {
    const int idx = blockIdx.x * 256 + threadIdx.x;
    if (idx >= BATCH * M_TOT) return;
    const int og = idx & (M_TOT - 1);
    const int o = og >> 3;
    const int g = og & 7;
    out[idx] = bred[g * OUT_F + o];
}

// ---------------------------------------------------------------------------
// Kernel 3: pack x into WMMA-B-fragment-ready layout:
//   xP[kpair*128 + n*2 + r] = x'[2*kpair + r, n] = x[n, i, c],  kk = 2*kpair+r
// so a lane's B pair (rows k+2h, k+2h+1 at column n) is one contiguous float2.
// One-time 1 MB repack; xP stays resident in the 192 MB L2.
// ---------------------------------------------------------------------------
__global__ __launch_bounds__(256)
void xpack_kernel(const float* __restrict__ x, float* __restrict__ xP) {
    const int idx = blockIdx.x * 256 + threadIdx.x;
    if (idx >= K_TOT * BATCH) return;
    const int kpair = idx >> 7;      // 0..2047
    const int j     = idx & 127;
    const int n     = j >> 1;        // batch column
    const int r     = j & 1;
    const int kk    = kpair * 2 + r;
    const int c     = kk >> 9;
    const int i     = kk & 511;
    xP[idx] = x[(size_t)n * (IN_F * IN_CH) + i * IN_CH + c];
}

// ---------------------------------------------------------------------------
// Kernel 4: split-K WMMA GEMM, fp32 via V_WMMA_F32_16X16X4_F32.
//   grid = (64 M-blocks, 8 K-chunks == one channel c each), block = 128.
//   Wave w owns M-tile Mbase = (blockIdx.x*4 + w)*16, all N=64, K = c's 512.
//   Constant-stride pointer walk (no per-step shift/mask), unroll 8.
//   A is single-use -> non-temporal; xP is hot in L2 -> regular loads.
//   Result: atomicAdd(acc/8) into out (native GLOBAL_ATOMIC_ADD_F32).
// A frag (16x4 f32, 2 VGPR): lanes 0-15 -> K {k,k+1}; lanes 16-31 -> {k+2,k+3}.
// B frag mirrored. C/D: lane l, vgpr p -> M = p + 8*(l>>4), N = l&15.
// ---------------------------------------------------------------------------
__global__ __launch_bounds__(128)
void gemm_wmma_kernel(const float* __restrict__ W,
                      const float* __restrict__ xP,
                      float* __restrict__ out) {
    const int lane = threadIdx.x & 31;
    const int wave = threadIdx.x >> 5;
    const int ml   = lane & 15;
    const int half = lane >> 4;

    const int Mbase = (blockIdx.x * 4 + wave) * 16;
    const int c     = blockIdx.y;          // this wave's K chunk = channel c

    v8f acc0 = {}, acc1 = {}, acc2 = {}, acc3 = {};

    // A: lane's row of W'[m, k] inside channel c, starting at its K half
    const float* Ap = W + (size_t)c * GOI + (size_t)(Mbase + ml) * IN_F + 2 * half;
    // B: packed fragments; kpair base for k = c*512 is c*256 (+half)
    const float* Bp = xP + ((size_t)c * (IN_F / 2) + half) * (2 * BATCH) + 2 * ml;

#pragma unroll 8
    for (int s = 0; s < IN_F / 4; ++s) {
        v2f a  = __builtin_nontemporal_load((const v2f*)Ap);
        v2f b0 = *(const v2f*)(Bp + 0);    // N-tile 0 (b = 0..15)
        v2f b1 = *(const v2f*)(Bp + 32);   // N-tile 1 (b = 16..31)
        v2f b2 = *(const v2f*)(Bp + 64);   // N-tile 2 (b = 32..47)
        v2f b3 = *(const v2f*)(Bp + 96);   // N-tile 3 (b = 48..63)

        acc0 = __builtin_amdgcn_wmma_f32_16x16x4_f32(false, a, false, b0,
                                                     (short)0, acc0, false, false);
        acc1 = __builtin_amdgcn_wmma_f32_16x16x4_f32(false, a, false, b1,
                                                     (short)0, acc1, false, false);
        acc2 = __builtin_amdgcn_wmma_f32_16x16x4_f32(false, a, false, b2,
                                                     (short)0, acc2, false, false);
        acc3 = __builtin_amdgcn_wmma_f32_16x16x4_f32(false, a, false, b3,
                                                     (short)0, acc3, false, false);

        Ap += 4;                 // next 4 K values of this row
        Bp += 4 * BATCH;         // next 2 kpairs
    }

    // Epilogue: out[b*4096 + o*8 + g] += acc/IN_CH  (b = nt*16 + ml)
    const float scale = 1.0f / (float)IN_CH;
#pragma unroll
    for (int p = 0; p < 8; ++p) {
        const int m = Mbase + p + 8 * half;
        const int g = m >> 9;
        const int o = m & 511;
        const size_t obase = (size_t)o * OUT_CH + g;
        atomicAdd(out + ((size_t)(0 * 16 + ml) * M_TOT + obase), acc0[p] * scale);
        atomicAdd(out + ((size_t)(1 * 16 + ml) * M_TOT + obase), acc1[p] * scale);
        atomicAdd(out + ((size_t)(2 * 16 + ml) * M_TOT + obase), acc2[p] * scale);
        atomicAdd(out + ((size_t)(3 * 16 + ml) * M_TOT + obase), acc3[p] * scale);
    }
}

// ---------------------------------------------------------------------------
// Host launcher
// ---------------------------------------------------------------------------
extern "C" void kernel_launch(void* const* d_in, const int* in_sizes, int n_in,
                              void* d_out, int out_size, void* d_ws, size_t ws_size,
                              hipStream_t stream) {
    const float* x    = (const float*)d_in[0];   // (64, 512, 8)
    const float* W    = (const float*)d_in[1];   // (8, 8, 512, 512)
    const float* bias = (const float*)d_in[2];   // (8, 8, 512, 512)
    float* out = (float*)d_out;                  // (64, 512, 8)

    float* bred = (float*)d_ws;                  // 4096 floats
    float* xP   = bred + M_TOT;                  // 4096*64 floats (1 MB)

    // 1. bias mean over (c, i)
    bias_reduce_kernel<<<M_TOT, 256, 0, stream>>>(bias, bred);

    // 2. seed output with bias term
    out_init_kernel<<<(BATCH * M_TOT + 255) / 256, 256, 0, stream>>>(bred, out);

    // 3. repack x -> xP (B-fragment-ready layout)
    xpack_kernel<<<(K_TOT * BATCH + 255) / 256, 256, 0, stream>>>(x, xP);

    // 4. split-K WMMA GEMM, accumulate into out
    dim3 grid(M_TOT / (4 * 16), IN_CH);  // 64 M-blocks x 8 channels
    gemm_wmma_kernel<<<grid, 128, 0, stream>>>(W, xP, out);
}